// Router_80676665688476
// MI455X (gfx1250) — compile-verified
//
#include <hip/hip_runtime.h>
#include <hip/hip_bf16.h>

// MoE router: logits = x[16384,2048] @ W[64,2048]^T, f32 softmax over 64 experts,
// top-8 -> dense gates [T,64] f32 + one-hot map [T,64] i32.
// Memory-bound (~150MB @ 23.3TB/s). fp32 WMMA 16x16x4 for the GEMM; x is
// streamed into double-buffered LDS by the Tensor Data Mover (one
// tensor_load_to_lds per 128x64 tile, hardware row padding), overlapping the
// DMA of chunk c+1 with WMMA consumption of chunk c.

typedef __attribute__((ext_vector_type(2))) float v2f;
typedef __attribute__((ext_vector_type(8))) float v8f;
typedef unsigned int v4u __attribute__((ext_vector_type(4)));
typedef int          v4i __attribute__((ext_vector_type(4)));
typedef int          v8i __attribute__((ext_vector_type(8)));

#define T_TOK   16384
#define H_DIM   2048
#define N_EXP   64
#define TOPK    8
#define BROWS   128            // token rows per block
#define BK      64             // K-chunk in floats (256B per row)
#define LSTRIDE (BK + 4)       // padded LDS row stride (68 floats)
#define NCHUNK  (H_DIM / BK)   // 32
#define NTHR    256
#define BUF_FLOATS (BROWS * LSTRIDE)

// One TDM descriptor: load a [BROWS x BK] f32 tile (row stride H_DIM elements)
// from global into LDS at lds_off, inserting 4 dwords of padding after every
// 64 dwords (pad_interval=5 -> 64 dwords, pad_amount=3 -> 4 dwords).
__device__ __forceinline__ void tdm_load_xtile(const float* gsrc, unsigned lds_off) {
  unsigned long long ga = (unsigned long long)(uintptr_t)gsrc;
  v4u g0 = { 1u,                                   // count=1 (valid descriptor)
             lds_off,                              // lds_addr [63:32]
             (unsigned)ga,                         // global_addr [95:64]
             (unsigned)((ga >> 32) & 0x01FFFFFFu)  // global_addr [120:96]
               | (2u << 30) };                     // type=2 ("image") [127:126]
  v8i g1 = { (int)((2u << 16)                      // data_size = 4B
               | (1u << 20)                        // pad_enable
               | (5u << 22)                        // pad_interval: 64 dwords
               | (3u << 25)),                      // pad_amount: 4 dwords
             (int)(((unsigned)H_DIM & 0xFFFFu) << 16),  // tensor_dim0[15:0]
             (int)(((unsigned)T_TOK & 0xFFFFu) << 16),  // dim0[31:16]=0 | tensor_dim1[15:0]
             (int)((unsigned)BK << 16),                 // dim1[31:16]=0 | tile_dim0=64
             (int)BROWS,                                // tile_dim1=128 | tile_dim2=0
             (int)H_DIM,                                // tensor_dim0_stride = 2048
             0, 0 };                                    // stride hi / tensor_dim1_stride
  v4i g2 = { 0, 0, 0, 0 };                              // 2-D tile: groups 2/3 unused
  v4i g3 = { 0, 0, 0, 0 };
  v8i g4 = { 0, 0, 0, 0, 0, 0, 0, 0 };                  // extra group (clang-23 form)
  __builtin_amdgcn_tensor_load_to_lds(g0, g1, g2, g3, g4, 0);
}

__global__ void __launch_bounds__(NTHR) router_kernel(
    const float* __restrict__ x, const float* __restrict__ w,
    float* __restrict__ gates, int* __restrict__ map) {
  __shared__ float lds[2 * BUF_FLOATS];

  const int tid  = threadIdx.x;
  const int wave = tid >> 5;
  const int lane = tid & 31;
  const int row0 = blockIdx.x * BROWS;

  const int m    = lane & 15;            // row-in-slab / expert-in-tile index
  const int koff = (lane >> 4) << 1;     // lanes 0-15 -> K0,K1 ; 16-31 -> K2,K3

  v8f acc0 = {}, acc1 = {}, acc2 = {}, acc3 = {};

  // Loop-invariant per-lane B base pointers (expert rows of W), 4 N-tiles.
  const float* wb0 = w + (size_t)(0 * 16 + m) * H_DIM + koff;
  const float* wb1 = w + (size_t)(1 * 16 + m) * H_DIM + koff;
  const float* wb2 = w + (size_t)(2 * 16 + m) * H_DIM + koff;
  const float* wb3 = w + (size_t)(3 * 16 + m) * H_DIM + koff;
  // Per-lane A base (float index) into each LDS buffer.
  const int abase = (wave * 16 + m) * LSTRIDE + koff;

  const float* xrow0 = x + (size_t)row0 * H_DIM;     // this block's tile origin
  const unsigned lds0 = (unsigned)(uintptr_t)&lds[0];          // LDS byte offset
  const unsigned bufbytes = BUF_FLOATS * 4u;

  // Prefetch chunk 0 into buffer 0 (single wave-level TDM op).
  if (wave == 0) tdm_load_xtile(xrow0, lds0);

  for (int c = 0; c < NCHUNK; ++c) {
    const int cur = c & 1;
    __builtin_amdgcn_s_wait_tensorcnt(0);   // wave0: chunk c landed; others: no-op
    __syncthreads();                        // all waves: chunk c visible, chunk c-1
                                            // buffer (1-cur) no longer being read
    if (wave == 0 && c + 1 < NCHUNK)
      tdm_load_xtile(xrow0 + (c + 1) * BK, lds0 + (unsigned)(1 - cur) * bufbytes);

    const int ab = abase + cur * BUF_FLOATS;
    const int kc = c * BK;
#pragma unroll
    for (int i = 0; i < BK / 4; ++i) {
      const int k = 4 * i;
      v2f a  = *(const v2f*)(&lds[ab + k]);
      v2f b0 = *(const v2f*)(wb0 + kc + k);
      v2f b1 = *(const v2f*)(wb1 + kc + k);
      v2f b2 = *(const v2f*)(wb2 + kc + k);
      v2f b3 = *(const v2f*)(wb3 + kc + k);
      acc0 = __builtin_amdgcn_wmma_f32_16x16x4_f32(false, a, false, b0, (short)0, acc0, false, false);
      acc1 = __builtin_amdgcn_wmma_f32_16x16x4_f32(false, a, false, b1, (short)0, acc1, false, false);
      acc2 = __builtin_amdgcn_wmma_f32_16x16x4_f32(false, a, false, b2, (short)0, acc2, false, false);
      acc3 = __builtin_amdgcn_wmma_f32_16x16x4_f32(false, a, false, b3, (short)0, acc3, false, false);
    }
  }

  // Spill logits into buffer 0 (dead: last chunk computed from buffer 1, and
  // the chunk-31 barrier proved everyone finished reading buffer 0).
  // Wave w writes only its own 16-row slab; epilogue threads 0-127 then read rows across slabs.
#pragma unroll
  for (int j = 0; j < 8; ++j) {
    int rl = j + ((lane >> 4) << 3);       // C/D layout: M = j + (lane<16 ? 0 : 8)
    float* dst = &lds[(wave * 16 + rl) * LSTRIDE];
    dst[0 * 16 + m] = acc0[j];
    dst[1 * 16 + m] = acc1[j];
    dst[2 * 16 + m] = acc2[j];
    dst[3 * 16 + m] = acc3[j];
  }
  __syncthreads();

  // Epilogue: one thread per row. Gates are full-softmax probs of the top-8
  // (NOT renormalized). Strict '>' argmax matches JAX lowest-index tie-break.
  if (tid < BROWS) {
    const float* Lr = &lds[tid * LSTRIDE];
    float p[N_EXP];
#pragma unroll
    for (int e = 0; e < N_EXP; ++e) p[e] = Lr[e];
    float mx = p[0];
#pragma unroll
    for (int e = 1; e < N_EXP; ++e) mx = fmaxf(mx, p[e]);
    float sum = 0.f;
#pragma unroll
    for (int e = 0; e < N_EXP; ++e) { p[e] = __expf(p[e] - mx); sum += p[e]; }
    const float inv = 1.0f / sum;

    unsigned mlo = 0u, mhi = 0u;
#pragma unroll
    for (int j = 0; j < TOPK; ++j) {
      float best = -1.0f;
      int bi = 0;
#pragma unroll
      for (int e = 0; e < N_EXP; ++e) {
        unsigned sel = (e < 32) ? (mlo >> e) : (mhi >> (e - 32));
        float v = (sel & 1u) ? -1.0f : p[e];
        if (v > best) { best = v; bi = e; }
      }
      if (bi < 32) mlo |= (1u << bi); else mhi |= (1u << (bi - 32));
    }

    const size_t orow = (size_t)(row0 + tid) * N_EXP;
#pragma unroll
    for (int e4 = 0; e4 < N_EXP / 4; ++e4) {
      float g[4];
      int   mm[4];
#pragma unroll
      for (int cidx = 0; cidx < 4; ++cidx) {
        int e = e4 * 4 + cidx;
        unsigned sel = ((e < 32) ? (mlo >> e) : (mhi >> (e - 32))) & 1u;
        g[cidx]  = sel ? p[e] * inv : 0.0f;
        mm[cidx] = (int)sel;
      }
      *(float4*)(gates + orow + e4 * 4) = make_float4(g[0], g[1], g[2], g[3]);
      *(int4*)(map + orow + e4 * 4)     = make_int4(mm[0], mm[1], mm[2], mm[3]);
    }
  }
}

extern "C" void kernel_launch(void* const* d_in, const int* in_sizes, int n_in,
                              void* d_out, int out_size, void* d_ws, size_t ws_size,
                              hipStream_t stream) {
  const float* x = (const float*)d_in[0];   // [16384, 2048] f32
  const float* w = (const float*)d_in[1];   // [64, 2048] f32
  float* gates = (float*)d_out;                                  // [16384*64] f32
  int*   map   = (int*)d_out + (size_t)T_TOK * N_EXP;            // [16384*64] i32
  router_kernel<<<T_TOK / BROWS, NTHR, 0, stream>>>(x, w, gates, map);
}